// BatchedValuemetric_40535901339902
// MI455X (gfx1250) — compile-verified
//
#include <hip/hip_runtime.h>

// Problem constants (fixed by setup_inputs): images (8, 3, 8, 256, 256) f32
constexpr int kN   = 8;
constexpr int kT   = 8;
constexpr int kH   = 256;
constexpr int kW   = 256;
constexpr int kHW  = kH * kW;        // 65536 pixels per frame
constexpr int kTHW = kT * kHW;       // 524288 = one channel plane per batch
constexpr int kCTHW = 3 * kTHW;      // 1572864 = one batch item

typedef __attribute__((ext_vector_type(2))) float v2f;
typedef __attribute__((ext_vector_type(8))) float v8f;

// ---------------------------------------------------------------------------
// Kernel 1: per-frame mean of grayscale ("contrast"), only for aug==1 frames.
// Wave-level reduction done with v_wmma_f32_16x16x4_f32 against an all-ones
// B matrix: with a.x = per-lane partial and a.y = 0, the ISA A layout gives
// D[M][N] = acc_M + acc_{M+16}, so lane0's and lane16's 8 accumulator VGPRs
// together hold the whole 32-lane sum.
// ---------------------------------------------------------------------------
__global__ __launch_bounds__(256) void contrast_mean_kernel(
    const float* __restrict__ img, const int* __restrict__ augs,
    float* __restrict__ contrast) {
  const int frame = blockIdx.x;          // 0..63  (= n*kT + t)
  const int n = frame / kT;
  const int t = frame % kT;

  if (augs[n] != 1) {                    // block-uniform: EXEC stays all-ones
    if (threadIdx.x == 0) contrast[frame] = 0.0f;
    return;
  }

  const float* rp = img + (size_t)n * kCTHW + (size_t)t * kHW;
  const float* gp = rp + kTHW;
  const float* bp = rp + 2 * kTHW;

  float acc = 0.0f;
  // 65536 px = 16384 float4; 256 threads -> 64 coalesced b128 iterations/chan
  for (int i = threadIdx.x; i < kHW / 4; i += 256) {
    float4 r = ((const float4*)rp)[i];
    float4 g = ((const float4*)gp)[i];
    float4 b = ((const float4*)bp)[i];
    acc += 0.2989f * r.x + 0.587f * g.x + 0.114f * b.x;
    acc += 0.2989f * r.y + 0.587f * g.y + 0.114f * b.y;
    acc += 0.2989f * r.z + 0.587f * g.z + 0.114f * b.z;
    acc += 0.2989f * r.w + 0.587f * g.w + 0.114f * b.w;
  }

  // --- wave32 reduction via WMMA (ones B-matrix row sums) ---
  v2f a;  a.x = acc;  a.y = 0.0f;        // A: 16x4, lanes hold (M, K-pair)
  v2f bm; bm.x = 1.0f; bm.y = 1.0f;      // B: 4x16 all ones
  v8f c = {};
  v8f d = __builtin_amdgcn_wmma_f32_16x16x4_f32(
      /*neg_a=*/false, a, /*neg_b=*/false, bm,
      /*c_mod=*/(short)0, c, /*reuse_a=*/false, /*reuse_b=*/false);
  float s = d[0] + d[1] + d[2] + d[3] + d[4] + d[5] + d[6] + d[7];
  // lane 0 holds rows 0..7 summed; lane 16 holds rows 8..15 summed
  float wave_sum = __shfl(s, 0) + __shfl(s, 16);

  __shared__ float warp_sums[8];
  const int lane = threadIdx.x & 31;
  const int wid  = threadIdx.x >> 5;
  if (lane == 0) warp_sums[wid] = wave_sum;
  __syncthreads();
  if (threadIdx.x == 0) {
    float tot = 0.0f;
    for (int i = 0; i < 8; ++i) tot += warp_sums[i];
    contrast[frame] = tot * (1.0f / (float)kHW);
  }
}

// ---------------------------------------------------------------------------
// HSV hue rotation for one pixel, replicating the JAX reference exactly.
// ---------------------------------------------------------------------------
__device__ __forceinline__ void hue_adjust(float r, float g, float b, float hue,
                                           float& ro, float& go, float& bo) {
  float maxc = fmaxf(r, fmaxf(g, b));
  float minc = fminf(r, fminf(g, b));
  bool  eq   = (maxc == minc);
  float cr   = maxc - minc;
  float s    = cr / (eq ? 1.0f : maxc);
  float crd  = eq ? 1.0f : cr;
  float rc   = (maxc - r) / crd;
  float gc   = (maxc - g) / crd;
  float bc   = (maxc - b) / crd;
  float hr = (maxc == r) ? (bc - gc) : 0.0f;
  float hg = ((maxc == g) && (maxc != r)) ? (2.0f + rc - bc) : 0.0f;
  float hb = ((maxc != g) && (maxc != r)) ? (4.0f + gc - rc) : 0.0f;
  // mod(mod(x,1)+hue,1) == mod(x+hue,1); jnp.mod(x,1) = x - floor(x)
  float hh = (hr + hg + hb) / 6.0f + 1.0f + hue;
  float h  = hh - floorf(hh);

  float six = h * 6.0f;
  float fi  = floorf(six);
  float f   = six - fi;
  int   i   = ((int)fi) % 6;           // h in [0,1) -> i in [0,5]
  float v   = maxc;
  float p  = fminf(fmaxf(v * (1.0f - s),               0.0f), 1.0f);
  float q  = fminf(fmaxf(v * (1.0f - s * f),           0.0f), 1.0f);
  float tt = fminf(fmaxf(v * (1.0f - s * (1.0f - f)),  0.0f), 1.0f);
  switch (i) {
    case 0:  ro = v;  go = tt; bo = p;  break;
    case 1:  ro = q;  go = v;  bo = p;  break;
    case 2:  ro = p;  go = v;  bo = tt; break;
    case 3:  ro = p;  go = q;  bo = v;  break;
    case 4:  ro = tt; go = p;  bo = v;  break;
    default: ro = v;  go = p;  bo = q;  break;
  }
}

// ---------------------------------------------------------------------------
// Kernel 2: fused augment + blend + clip. One float4 (4 pixels x 3 channels)
// per thread; n (and thus the aug branch) is uniform per block.
// ---------------------------------------------------------------------------
__global__ __launch_bounds__(256) void augment_kernel(
    const float* __restrict__ img, const int* __restrict__ augs,
    const float* __restrict__ hue_f, const float* __restrict__ blend_f,
    const float* __restrict__ contrast, float* __restrict__ out) {
  constexpr int kBlocksPerN = (kTHW / 4) / 256;  // 512
  const int n    = blockIdx.x / kBlocksPerN;     // uniform per block
  const int idx4 = (blockIdx.x % kBlocksPerN) * 256 + threadIdx.x;
  const int pix  = idx4 * 4;                     // pixel index within plane
  const int t    = pix >> 16;                    // pix / kHW (4 px share t)

  const int   aug = augs[n];
  const float hue = hue_f[n];
  const float f   = (aug == 2 || aug == 3 || aug == 4) ? 0.0f : blend_f[n];
  const float ctr = (aug == 1) ? contrast[n * kT + t] : 0.0f;

  const size_t base = (size_t)n * kCTHW;
  float4 r4 = *(const float4*)(img + base + pix);
  float4 g4 = *(const float4*)(img + base + kTHW + pix);
  float4 b4 = *(const float4*)(img + base + 2 * kTHW + pix);

  float R[4] = {r4.x, r4.y, r4.z, r4.w};
  float G[4] = {g4.x, g4.y, g4.z, g4.w};
  float B[4] = {b4.x, b4.y, b4.z, b4.w};
  float OR[4], OG[4], OB[4];

  const float omf = 1.0f - f;
#pragma unroll
  for (int k = 0; k < 4; ++k) {
    float br, bg, bb;
    if (aug == 0) {
      br = bg = bb = 0.0f;
    } else if (aug == 1) {
      br = bg = bb = ctr;
    } else if (aug == 2 || aug == 5) {
      float gray = 0.2989f * R[k] + 0.587f * G[k] + 0.114f * B[k];
      br = bg = bb = gray;
    } else if (aug == 3) {
      hue_adjust(R[k], G[k], B[k], hue, br, bg, bb);
    } else {  // aug == 4
      br = R[k]; bg = G[k]; bb = B[k];
    }
    OR[k] = fminf(fmaxf(f * R[k] + omf * br, 0.0f), 1.0f);
    OG[k] = fminf(fmaxf(f * G[k] + omf * bg, 0.0f), 1.0f);
    OB[k] = fminf(fmaxf(f * B[k] + omf * bb, 0.0f), 1.0f);
  }

  *(float4*)(out + base + pix)            = make_float4(OR[0], OR[1], OR[2], OR[3]);
  *(float4*)(out + base + kTHW + pix)     = make_float4(OG[0], OG[1], OG[2], OG[3]);
  *(float4*)(out + base + 2 * kTHW + pix) = make_float4(OB[0], OB[1], OB[2], OB[3]);
}

// ---------------------------------------------------------------------------
extern "C" void kernel_launch(void* const* d_in, const int* in_sizes, int n_in,
                              void* d_out, int out_size, void* d_ws, size_t ws_size,
                              hipStream_t stream) {
  const float* images   = (const float*)d_in[0];
  const int*   sel_augs = (const int*)d_in[1];
  const float* hue_fs   = (const float*)d_in[2];
  const float* blend_fs = (const float*)d_in[3];
  float*       contrast = (float*)d_ws;          // kN*kT = 64 floats
  float*       outp     = (float*)d_out;

  // Pass 1: per-frame contrast means (only touches aug==1 batches; also
  // warms L2 — the whole 50 MB input fits in the 192 MB L2).
  contrast_mean_kernel<<<kN * kT, 256, 0, stream>>>(images, sel_augs, contrast);

  // Pass 2: fused augmentation, one float4 triple per thread.
  constexpr int total4 = (kN * kTHW) / 4;        // 1,048,576
  augment_kernel<<<total4 / 256, 256, 0, stream>>>(images, sel_augs, hue_fs,
                                                   blend_fs, contrast, outp);
}